// VmfVectorQuantizer_18116172054715
// MI455X (gfx1250) — compile-verified
//
#include <hip/hip_runtime.h>
#include <hip/hip_bf16.h>

typedef __attribute__((ext_vector_type(2))) float v2f;
typedef __attribute__((ext_vector_type(8))) float v8f;

#define DIMD 64
#define KCODES 512
#define NROWS 65536
#define TILE_ROWS 16

__device__ __forceinline__ float wave_sum(float v) {
#pragma unroll
  for (int off = 16; off > 0; off >>= 1) v += __shfl_xor(v, off, 32);
  return v;
}
__device__ __forceinline__ float wave_max(float v) {
#pragma unroll
  for (int off = 16; off > 0; off >>= 1) v = fmaxf(v, __shfl_xor(v, off, 32));
  return v;
}

// ---------------------------------------------------------------- zero accum
__global__ void vq_zero(float* acc) {
  int i = blockIdx.x * blockDim.x + threadIdx.x;
  if (i < 514) acc[i] = 0.0f;
}

// ---------------------------------------------------- codebook L2 normalize
__global__ __launch_bounds__(256) void vq_cbnorm(const float* __restrict__ cb,
                                                 float* __restrict__ cbn) {
  int wave = (blockIdx.x * blockDim.x + threadIdx.x) >> 5;  // 0..511 (one row per wave)
  int lane = threadIdx.x & 31;
  const float* row = cb + wave * DIMD;
  float v0 = row[lane], v1 = row[lane + 32];
  float s = wave_sum(v0 * v0 + v1 * v1);
  float scale = 1.0f / fmaxf(sqrtf(s), 1e-12f);
  cbn[wave * DIMD + lane] = v0 * scale;
  cbn[wave * DIMD + lane + 32] = v1 * scale;
}

// ------------------------------------------------------------- main kernel
// 1 wave32 per block, 16 rows per block, grid = 4096
__global__ __launch_bounds__(32) void vq_main(
    const float* __restrict__ z, const float* __restrict__ kappa_p,
    const float* __restrict__ cbn, const float* __restrict__ gum,
    float* __restrict__ out, float* __restrict__ avgp_acc,
    float* __restrict__ sdisc, float* __restrict__ scont) {
  __shared__ float zt[TILE_ROWS][DIMD];      // normalized z tile
  __shared__ float ybuf[TILE_ROWS][KCODES];  // logits, then encodings (in place)
  __shared__ float avgp[KCODES];             // per-block probability accumulation

  const int lane = threadIdx.x;
  const int m = lane & 15;
  const int hi = lane >> 4;
  const long base_row = (long)blockIdx.x * TILE_ROWS;
  const float kappa = kappa_p[0];

#pragma unroll
  for (int i = lane; i < KCODES; i += 32) avgp[i] = 0.0f;

  // ---- load + L2-normalize 16 rows of z into LDS
#pragma unroll
  for (int r = 0; r < TILE_ROWS; ++r) {
    const float* zr = z + (base_row + r) * DIMD;
    float v0 = zr[lane], v1 = zr[lane + 32];
    float s = wave_sum(v0 * v0 + v1 * v1);
    float scale = 1.0f / fmaxf(sqrtf(s), 1e-12f);
    zt[r][lane] = v0 * scale;
    zt[r][lane + 32] = v1 * scale;
  }
  __syncthreads();

  // ---- GEMM1: logits[16][512] = z_norm @ cb_norm^T  (f32 WMMA, K=64 in 16 steps)
  for (int ct = 0; ct < KCODES / 16; ++ct) {
    v8f c = {};
    const float* cbrow = cbn + (ct * 16 + m) * DIMD;
#pragma unroll
    for (int k = 0; k < 16; ++k) {
      int kb = 4 * k + 2 * hi;
      v2f a;
      a.x = zt[m][kb];
      a.y = zt[m][kb + 1];
      v2f b;
      b.x = cbrow[kb];
      b.y = cbrow[kb + 1];
      c = __builtin_amdgcn_wmma_f32_16x16x4_f32(false, a, false, b, (short)0,
                                                c, false, false);
    }
#pragma unroll
    for (int j = 0; j < 8; ++j)
      ybuf[j + 8 * hi][ct * 16 + m] = kappa * c[j];  // D: row j+8*hi, col m
  }
  __syncthreads();

  // ---- per-row dual softmax (log_softmax stats + gumbel-softmax encodings)
  // lane owns 16 CONTIGUOUS columns [lane*16, lane*16+16): gumbel reads become
  // four b128 loads per row (dominant 128MB stream), fully coalesced wave-wide.
  float acc_pl = 0.0f;
  for (int r = 0; r < TILE_ROWS; ++r) {
    const float4* urow4 =
        (const float4*)(gum + (base_row + r) * (long)KCODES) + lane * 4;
    float yv[16], y2v[16];
    float m1 = -3.4e38f, m2 = -3.4e38f;
#pragma unroll
    for (int i4 = 0; i4 < 4; ++i4) {
      float4 u4 = urow4[i4];
      float uu[4] = {u4.x, u4.y, u4.z, u4.w};
#pragma unroll
      for (int j = 0; j < 4; ++j) {
        int i = i4 * 4 + j;
        int c0 = lane * 16 + i;
        float yy = ybuf[r][c0];
        float g = -logf(-logf(uu[j] + 1e-10f) + 1e-10f);
        float y2 = (yy + g) * 2.0f;  // /TEMPERATURE (0.5)
        yv[i] = yy;
        y2v[i] = y2;
        m1 = fmaxf(m1, yy);
        m2 = fmaxf(m2, y2);
      }
    }
    m1 = wave_max(m1);
    m2 = wave_max(m2);
    float s1 = 0.0f, s2 = 0.0f;
#pragma unroll
    for (int i = 0; i < 16; ++i) {
      s1 += expf(yv[i] - m1);
      s2 += expf(y2v[i] - m2);
    }
    s1 = wave_sum(s1);
    s2 = wave_sum(s2);
    float lse = m1 + logf(s1);
    float inv_s2 = 1.0f / s2;
#pragma unroll
    for (int i = 0; i < 16; ++i) {
      int c0 = lane * 16 + i;
      float lp = yv[i] - lse;
      float p = expf(lp);
      acc_pl += p * lp;
      avgp[c0] += p;  // lane owns contiguous column block: race-free
      ybuf[r][c0] = expf(y2v[i] - m2) * inv_s2;  // encodings in place
    }
  }
  acc_pl = wave_sum(acc_pl);
  if (lane == 0) atomicAdd(sdisc, acc_pl);
#pragma unroll
  for (int i = lane; i < KCODES; i += 32) atomicAdd(&avgp_acc[i], avgp[i]);
  __syncthreads();

  // ---- GEMM2: z_q[16][64] = encodings[16][512] @ cb_norm[512][64]
  // fused with continuous-KL reduction and output store
  float acc_cont = 0.0f;
  for (int nt = 0; nt < 4; ++nt) {
    v8f c = {};
    for (int k = 0; k < KCODES / 4; ++k) {
      int kb = 4 * k + 2 * hi;
      v2f a;
      a.x = ybuf[m][kb];
      a.y = ybuf[m][kb + 1];
      v2f b;
      b.x = cbn[(long)kb * DIMD + nt * 16 + m];
      b.y = cbn[(long)(kb + 1) * DIMD + nt * 16 + m];
      c = __builtin_amdgcn_wmma_f32_16x16x4_f32(false, a, false, b, (short)0,
                                                c, false, false);
    }
#pragma unroll
    for (int j = 0; j < 8; ++j) {
      int r = j + 8 * hi;
      long idx = (base_row + r) * DIMD + nt * 16 + m;
      float zq = c[j];
      float zo = z[idx];
      acc_cont += zo * (zo - zq);
      out[idx] = zq;
    }
  }
  acc_cont = wave_sum(acc_cont);
  if (lane == 0) atomicAdd(scont, acc_cont);
}

// ------------------------------------------------------------- finalize
__global__ __launch_bounds__(32) void vq_finalize(
    const float* __restrict__ kappa_p, const float* __restrict__ avgp_acc,
    const float* __restrict__ sdisc, const float* __restrict__ scont,
    float* __restrict__ out) {
  int lane = threadIdx.x;
  const float invN = 1.0f / (float)NROWS;
  float acc = 0.0f;
#pragma unroll
  for (int i = lane; i < KCODES; i += 32) {
    float a = avgp_acc[i] * invN;
    acc += a * logf(a + 1e-7f);
  }
  acc = wave_sum(acc);
  if (lane == 0) {
    float loss = (sdisc[0] + kappa_p[0] * scont[0]) / 8.0f;
    out[(long)NROWS * DIMD] = loss;            // loss after z_to_decoder
    out[(long)NROWS * DIMD + 1] = expf(-acc);  // perplexity
  }
}

extern "C" void kernel_launch(void* const* d_in, const int* in_sizes, int n_in,
                              void* d_out, int out_size, void* d_ws,
                              size_t ws_size, hipStream_t stream) {
  const float* z = (const float*)d_in[0];      // [8,1024,512]
  const float* kappa = (const float*)d_in[1];  // scalar
  const float* cb = (const float*)d_in[2];     // [512,64]
  const float* gum = (const float*)d_in[3];    // [65536,512]
  float* out = (float*)d_out;                  // [65536*64 + 2]

  float* wsf = (float*)d_ws;
  float* avgp_acc = wsf;     // [512]
  float* sdisc = wsf + 512;  // [1]
  float* scont = wsf + 513;  // [1]
  float* cbn = wsf + 1024;   // [512*64] normalized codebook

  vq_zero<<<3, 256, 0, stream>>>(wsf);
  vq_cbnorm<<<64, 256, 0, stream>>>(cb, cbn);
  vq_main<<<NROWS / TILE_ROWS, 32, 0, stream>>>(z, kappa, cbn, gum, out,
                                                avgp_acc, sdisc, scont);
  vq_finalize<<<1, 32, 0, stream>>>(kappa, avgp_acc, sdisc, scont, out);
}